// Mamba_60284160967079
// MI455X (gfx1250) — compile-verified
//
#include <hip/hip_runtime.h>
#include <hip/hip_bf16.h>

// ---------------------------------------------------------------------------
// Mamba SSM block for gfx1250 (MI455X, wave32, WMMA).
// All matrix work is mapped onto V_WMMA_F32_16X16X4_F32 (f32, 16x16 C, K=4).
// Fragment layouts per CDNA5 ISA 7.12.2:
//   A-frag (16x4 f32, 2 VGPR/lane): lane l -> row (l&15), k = 4q + 2*(l>>4) .. +1
//   B-frag (4x16 f32, 2 VGPR/lane): lane l -> col (l&15), rows k, k+1 (same k map)
//   C-frag (16x16 f32, 8 VGPR/lane): vgpr r -> row r + 8*(l>>4), col (l&15)
// The sequential scan double-buffers its per-timestep matrices in LDS via
// GLOBAL_LOAD_ASYNC_TO_LDS_B128 + S_WAIT_ASYNCCNT (CDNA5 async copy path).
// ---------------------------------------------------------------------------

typedef __attribute__((ext_vector_type(2))) float v2f;
typedef __attribute__((ext_vector_type(8))) float v8f;

#define DMODEL 512
#define DCONV  4
#define DSTATE 16
#define DTOT   1024
#define BATCH  2
#define LSEQ   2048
#define NPOS   (BATCH * LSEQ)          // 4096
#define NS     (1 + DSTATE + 2 * DTOT) // 2065

__device__ __forceinline__ float softplusf(float x) {
    return (x > 20.0f) ? x : log1pf(expf(x));
}
__device__ __forceinline__ float sigmoidf(float x) {
    return 1.0f / (1.0f + expf(-x));
}

#define WMMA_F32(a, b, c) \
    __builtin_amdgcn_wmma_f32_16x16x4_f32(false, (a), false, (b), (short)0, (c), false, false)

// Wave-relative LDS byte address = low 32 bits of the flat pointer (ISA 10.2).
__device__ __forceinline__ unsigned lds_addr(const void* p) {
    return (unsigned)(unsigned long long)p;
}
// Async copy global->LDS, 16B per lane, GVS addressing (SGPR base + VGPR off).
__device__ __forceinline__ void async_to_lds_b128(unsigned ldsoff,
                                                  const float* gbase,
                                                  unsigned goff) {
    asm volatile("global_load_async_to_lds_b128 %0, %1, %2"
                 :: "v"(ldsoff), "v"(goff), "s"(gbase) : "memory");
}
__device__ __forceinline__ void wait_asynccnt_le4() {
    asm volatile("s_wait_asynccnt 0x4" ::: "memory");
}

// ---------------------------------------------------------------------------
// f32 WMMA GEMM, 32x32 output tile per wave (2x2 C-fragments).
// C[m,n] = sum_k A[m,k]*B[k,n] + bias[n] (+ residual[m,n]).
// A-frags are reused across 2 B-frags and vice versa: 6 loads per 4 WMMAs.
// M = gridDim.y*32 (all our Ms are 4096). N may be ragged (W_s: N=2065):
// loads clamp the column, stores are guarded.
// ---------------------------------------------------------------------------
__global__ __launch_bounds__(32) void gemm32_wmma(
    const float* __restrict__ A, const float* __restrict__ B,
    const float* __restrict__ bias, const float* __restrict__ residual,
    float* __restrict__ C, int N, int K, int lda, int ldb, int ldc)
{
    const int lane = threadIdx.x;
    const int hi   = lane >> 4;
    const int lq   = lane & 15;
    const int m0   = blockIdx.y << 5;
    const int n0   = blockIdx.x << 5;

    const int col0 = n0 + lq;
    const int col1 = n0 + 16 + lq;
    const int c0   = (col0 < N) ? col0 : (N - 1);
    const int c1   = (col1 < N) ? col1 : (N - 1);

    const float* Arow0 = A + (size_t)(m0 + lq) * lda;
    const float* Arow1 = A + (size_t)(m0 + 16 + lq) * lda;

    v8f acc00 = {}, acc01 = {}, acc10 = {}, acc11 = {};
    for (int k = 0; k < K; k += 4) {
        const int ka = k + (hi << 1);
        v2f a0 = *(const v2f*)(Arow0 + ka);
        v2f a1 = *(const v2f*)(Arow1 + ka);
        const float* Bk0 = B + (size_t)ka * ldb;
        const float* Bk1 = B + (size_t)(ka + 1) * ldb;
        v2f b0; b0.x = Bk0[c0]; b0.y = Bk1[c0];
        v2f b1; b1.x = Bk0[c1]; b1.y = Bk1[c1];
        acc00 = WMMA_F32(a0, b0, acc00);
        acc01 = WMMA_F32(a0, b1, acc01);
        acc10 = WMMA_F32(a1, b0, acc10);
        acc11 = WMMA_F32(a1, b1, acc11);
    }

    auto store_tile = [&](const v8f& acc, int mbase, int col) {
        if (col < N) {
            const float bv = bias ? bias[col] : 0.0f;
            #pragma unroll
            for (int r = 0; r < 8; ++r) {
                const int m = mbase + (hi << 3) + r;
                float v = acc[r] + bv;
                if (residual) v += residual[(size_t)m * ldc + col];
                C[(size_t)m * ldc + col] = v;
            }
        }
    };
    store_tile(acc00, m0,      col0);
    store_tile(acc01, m0,      col1);
    store_tile(acc10, m0 + 16, col0);
    store_tile(acc11, m0 + 16, col1);
}

// ---------------------------------------------------------------------------
// Depthwise causal conv (DC=4, groups=D) + bias.
// ---------------------------------------------------------------------------
__global__ void conv_kernel(const float* __restrict__ u0,
                            const float* __restrict__ w,
                            const float* __restrict__ cb,
                            float* __restrict__ uc)
{
    const int idx = blockIdx.x * blockDim.x + threadIdx.x;
    if (idx >= NPOS * DTOT) return;
    const int d  = idx & (DTOT - 1);
    const int bl = idx >> 10;              // b*L + l
    const int l  = bl & (LSEQ - 1);
    float acc = cb[d];
    #pragma unroll
    for (int t = 0; t < DCONV; ++t) {
        const int ls = l - (DCONV - 1) + t;
        if (ls >= 0)
            acc += w[t * DTOT + d] * u0[(size_t)(bl - (DCONV - 1) + t) * DTOT + d];
    }
    uc[idx] = acc;
}

// ---------------------------------------------------------------------------
// 16x16 matrix inverse via Gauss-Jordan, single workgroup (256 threads).
// A = -(I + 0.1*N) is well-conditioned; no pivoting needed.
// ---------------------------------------------------------------------------
__global__ __launch_bounds__(256) void inv16_kernel(const float* __restrict__ A,
                                                    float* __restrict__ invA)
{
    __shared__ float M[16][33];            // padded to dodge bank conflicts
    const int tid = threadIdx.x;
    const int r = tid >> 4;
    const int c = tid & 15;
    M[r][c]      = A[r * 16 + c];
    M[r][c + 16] = (r == c) ? 1.0f : 0.0f;
    __syncthreads();
    for (int k = 0; k < 16; ++k) {
        const float pv = M[k][k];
        __syncthreads();
        if (r == k) {
            const float ip = 1.0f / pv;
            M[k][c] *= ip;
            M[k][c + 16] *= ip;
        }
        __syncthreads();
        const float fk = M[r][k];
        const float e0 = M[k][c];
        const float e1 = M[k][c + 16];
        __syncthreads();
        if (r != k) {
            M[r][c]      -= fk * e0;
            M[r][c + 16] -= fk * e1;
        }
        __syncthreads();
    }
    invA[r * 16 + c] = M[r][c + 16];
}

// ---------------------------------------------------------------------------
// Per-position discretization: A_disc = expm(delta*A) (scaling & squaring,
// Taylor-12 Horner), P = (A_disc - I) @ invA, A_scaled = A_disc * sigmoid rows.
// One wave per (b,l); Ei lives in LDS column-major so B-fragments are
// contiguous float2 loads. Single wave => LDS in-order, no barriers needed.
// Outputs stored ROW-major so the scan's B-fragments (transpose operand)
// load contiguously.
// ---------------------------------------------------------------------------
__global__ __launch_bounds__(32) void expm_kernel(
    const float* __restrict__ s, const float* __restrict__ Amat,
    const float* __restrict__ invA,
    float* __restrict__ Asc, float* __restrict__ P)
{
    const int pos  = blockIdx.x;
    const int lane = threadIdx.x;
    const int hi   = lane >> 4;
    const int lq   = lane & 15;
    const float* srow = s + (size_t)pos * NS;

    __shared__ float Ei[256];              // col-major: [col*16 + row]

    const float delta = softplusf(srow[0]);
    const float sc = delta * (1.0f / 16.0f);

    // Ms A-fragments (constant across Horner iterations)
    v2f af[4];
    #pragma unroll
    for (int q = 0; q < 4; ++q) {
        const int ka = 4 * q + 2 * hi;
        v2f t = *(const v2f*)(Amat + lq * 16 + ka);
        af[q] = t * sc;
    }

    // Ei = I
    #pragma unroll
    for (int r = 0; r < 8; ++r)
        Ei[lq * 16 + hi * 8 + r] = ((hi * 8 + r) == lq) ? 1.0f : 0.0f;

    // Horner: Ei = I + (Ms @ Ei) / k, k = 12..1
    for (int k = 12; k >= 1; --k) {
        v8f c = {};
        #pragma unroll
        for (int q = 0; q < 4; ++q) {
            v2f b = *(const v2f*)&Ei[lq * 16 + 4 * q + 2 * hi];
            c = WMMA_F32(af[q], b, c);
        }
        const float ik = 1.0f / (float)k;
        #pragma unroll
        for (int r = 0; r < 8; ++r) {
            const int rowm = hi * 8 + r;
            Ei[lq * 16 + rowm] = c[r] * ik + ((rowm == lq) ? 1.0f : 0.0f);
        }
    }

    // 4 squarings: Ei = Ei @ Ei
    for (int it = 0; it < 4; ++it) {
        v8f c = {};
        #pragma unroll
        for (int q = 0; q < 4; ++q) {
            const int ka = 4 * q + 2 * hi;
            v2f a;
            a.x = Ei[(ka + 0) * 16 + lq];
            a.y = Ei[(ka + 1) * 16 + lq];
            v2f b = *(const v2f*)&Ei[lq * 16 + ka];
            c = WMMA_F32(a, b, c);
        }
        #pragma unroll
        for (int r = 0; r < 8; ++r)
            Ei[lq * 16 + hi * 8 + r] = c[r];
    }

    // P = (Ei - I) @ invA, stored row-major
    {
        v8f p = {};
        #pragma unroll
        for (int q = 0; q < 4; ++q) {
            const int ka = 4 * q + 2 * hi;
            v2f a;
            a.x = Ei[(ka + 0) * 16 + lq] - ((lq == ka + 0) ? 1.0f : 0.0f);
            a.y = Ei[(ka + 1) * 16 + lq] - ((lq == ka + 1) ? 1.0f : 0.0f);
            v2f b;
            b.x = invA[(ka + 0) * 16 + lq];
            b.y = invA[(ka + 1) * 16 + lq];
            p = WMMA_F32(a, b, p);
        }
        float* Pp = P + (size_t)pos * 256;
        #pragma unroll
        for (int r = 0; r < 8; ++r)
            Pp[(hi * 8 + r) * 16 + lq] = p[r];
    }

    // A_scaled[i][j] = Ei[i][j] * sigmoid(s[1+i]), stored row-major
    {
        float* Ap = Asc + (size_t)pos * 256;
        #pragma unroll
        for (int r = 0; r < 8; ++r) {
            const int rowm = hi * 8 + r;
            const float sg = sigmoidf(srow[1 + rowm]);
            Ap[rowm * 16 + lq] = Ei[lq * 16 + rowm] * sg;
        }
    }
}

// ---------------------------------------------------------------------------
// Sequential scan. One wave per (batch, 16-channel tile): 2*64 = 128 waves.
// h (16 channels x 16 states) held in LDS row-major. Per step:
//   h <- h @ A_t^T  +  (u ∘ Bp) @ P^T  +  (u*Bmod) broadcast over states
//   y[d] = Cmod[d] * <h[d,:], Cp[d,:]> + Dp[d]*u[d]
// The 2x1KB per-step matrices are double-buffered in LDS: while timestep t is
// consumed (ds_load_b64 fragments -> 8 WMMAs), timestep t+1 streams in via
// GLOBAL_LOAD_ASYNC_TO_LDS_B128 (4 async ops/step, 16B/lane each), gated by
// S_WAIT_ASYNCCNT <= 4.
// ---------------------------------------------------------------------------
__global__ __launch_bounds__(32) void scan_kernel(
    const float* __restrict__ Asc, const float* __restrict__ P,
    const float* __restrict__ uc,  const float* __restrict__ s,
    const float* __restrict__ Bp,  const float* __restrict__ Cp,
    const float* __restrict__ Dp,  float* __restrict__ y)
{
    const int b    = blockIdx.y;
    const int d0   = blockIdx.x << 4;
    const int lane = threadIdx.x;
    const int hi   = lane >> 4;
    const int lq   = lane & 15;

    __shared__ __align__(16) float h[256];          // row-major: [drow*16 + state]
    __shared__ __align__(16) float mats[2][512];    // [buf][ A(256) | P(256) ]

    #pragma unroll
    for (int r = 0; r < 8; ++r) h[lane * 8 + r] = 0.0f;

    // B_param A-fragments (constant over time; scaled by u each step)
    v2f bpq[4];
    #pragma unroll
    for (int q = 0; q < 4; ++q)
        bpq[q] = *(const v2f*)(Bp + (size_t)(d0 + lq) * 16 + 4 * q + 2 * hi);

    // C_param row + D_param (used by lanes 0..15, d = d0+lane)
    float4 cp0, cp1, cp2, cp3;
    {
        const float4* cpr = (const float4*)(Cp + (size_t)(d0 + lq) * 16);
        cp0 = cpr[0]; cp1 = cpr[1]; cp2 = cpr[2]; cp3 = cpr[3];
    }
    const float dpar = Dp[d0 + lq];

    const unsigned lane16 = (unsigned)(lane * 16);

    // stage timestep t's A_scaled and P into mats[buf] (4 async b128 ops)
    auto stage = [&](int t, int buf) {
        const int tc = (t < LSEQ) ? t : (LSEQ - 1);
        const unsigned posoff = (unsigned)(((unsigned)b * LSEQ + tc) * 1024u); // bytes
        const unsigned la = lds_addr(&mats[buf][0]);
        const unsigned lp = lds_addr(&mats[buf][256]);
        async_to_lds_b128(la + lane16,       Asc, posoff + lane16);
        async_to_lds_b128(la + 512 + lane16, Asc, posoff + 512 + lane16);
        async_to_lds_b128(lp + lane16,       P,   posoff + lane16);
        async_to_lds_b128(lp + 512 + lane16, P,   posoff + 512 + lane16);
    };

    stage(0, 0);

    for (int t = 0; t < LSEQ; ++t) {
        const int buf = t & 1;
        stage(t + 1, buf ^ 1);          // prefetch next step into other buffer
        wait_asynccnt_le4();            // current buffer's 4 transfers retired

        const size_t pos = (size_t)b * LSEQ + t;
        const float* As   = &mats[buf][0];
        const float* Pm   = &mats[buf][256];
        const float* urow = uc + pos * DTOT + d0;
        const float* srow = s  + pos * NS;

        const float  u_scale = urow[lq];                 // u[d0 + (lane&15)]
        const float4 ua = *(const float4*)(urow + hi * 8);
        const float4 ub = *(const float4*)(urow + hi * 8 + 4);
        float bm8[8];
        #pragma unroll
        for (int r = 0; r < 8; ++r)
            bm8[r] = srow[1 + DSTATE + d0 + hi * 8 + r]; // B_mod

        v8f c = {};
        // h @ A_t^T : B[k][n] = A_scaled[n][k] -> contiguous in row-major tile
        #pragma unroll
        for (int q = 0; q < 4; ++q) {
            const int ka = 4 * q + 2 * hi;
            v2f ah = *(const v2f*)&h[lq * 16 + ka];
            v2f bA = *(const v2f*)&As[lq * 16 + ka];
            c = WMMA_F32(ah, bA, c);
        }
        // (u ∘ Bp) @ P^T
        #pragma unroll
        for (int q = 0; q < 4; ++q) {
            const int ka = 4 * q + 2 * hi;
            v2f bP = *(const v2f*)&Pm[lq * 16 + ka];
            v2f au = bpq[q] * u_scale;
            c = WMMA_F32(au, bP, c);
        }

        const float uu[8] = {ua.x, ua.y, ua.z, ua.w, ub.x, ub.y, ub.z, ub.w};
        #pragma unroll
        for (int r = 0; r < 8; ++r)
            h[(hi * 8 + r) * 16 + lq] = c[r] + uu[r] * bm8[r];

        // y output: lanes 0..15 each own channel d = d0+lane
        if (lane < 16) {
            const float4* hr = (const float4*)&h[lane * 16];
            const float4 h0 = hr[0], h1 = hr[1], h2 = hr[2], h3 = hr[3];
            float acc = h0.x * cp0.x + h0.y * cp0.y + h0.z * cp0.z + h0.w * cp0.w
                      + h1.x * cp1.x + h1.y * cp1.y + h1.z * cp1.z + h1.w * cp1.w
                      + h2.x * cp2.x + h2.y * cp2.y + h2.z * cp2.z + h2.w * cp2.w
                      + h3.x * cp3.x + h3.y * cp3.y + h3.z * cp3.z + h3.w * cp3.w;
            const float cmod = srow[1 + DSTATE + DTOT + d0 + lane];
            y[pos * DTOT + d0 + lane] = cmod * acc + dpar * u_scale;
        }
    }
}

// ---------------------------------------------------------------------------
// Host side
// ---------------------------------------------------------------------------
extern "C" void kernel_launch(void* const* d_in, const int* in_sizes, int n_in,
                              void* d_out, int out_size, void* d_ws, size_t ws_size,
                              hipStream_t stream)
{
    (void)in_sizes; (void)n_in; (void)out_size; (void)ws_size;

    const float* x       = (const float*)d_in[0];
    const float* W_in    = (const float*)d_in[1];
    const float* b_in    = (const float*)d_in[2];
    const float* conv_w  = (const float*)d_in[3];
    const float* conv_b  = (const float*)d_in[4];
    const float* A       = (const float*)d_in[5];
    const float* B_param = (const float*)d_in[6];
    const float* C_param = (const float*)d_in[7];
    const float* D_param = (const float*)d_in[8];
    const float* W_s     = (const float*)d_in[9];
    const float* b_s     = (const float*)d_in[10];
    const float* W_out   = (const float*)d_in[11];
    const float* b_out   = (const float*)d_in[12];
    float* out = (float*)d_out;

    // workspace carve-up (floats): ~93 MB total; fits in the 192 MB L2, so
    // intermediates stay L2-resident between kernels.
    float* ws   = (float*)d_ws;
    float* u0   = ws;                                  // NPOS*DTOT
    float* ucv  = u0  + (size_t)NPOS * DTOT;           // NPOS*DTOT
    float* sbuf = ucv + (size_t)NPOS * DTOT;           // NPOS*NS
    float* invA = sbuf + (size_t)NPOS * NS;            // 256
    float* Ascl = invA + 256;                          // NPOS*256
    float* Pmat = Ascl + (size_t)NPOS * 256;           // NPOS*256
    float* ybuf = Pmat + (size_t)NPOS * 256;           // NPOS*DTOT

    // 1) u = x @ W_in + b_in            (4096 x 512 x 1024)
    gemm32_wmma<<<dim3(DTOT / 32, NPOS / 32), 32, 0, stream>>>(
        x, W_in, b_in, nullptr, u0, DTOT, DMODEL, DMODEL, DTOT, DTOT);

    // 2) depthwise causal conv + bias
    conv_kernel<<<(NPOS * DTOT + 255) / 256, 256, 0, stream>>>(u0, conv_w, conv_b, ucv);

    // 3) s = u @ W_s + b_s              (4096 x 1024 x 2065, ragged N)
    gemm32_wmma<<<dim3((NS + 31) / 32, NPOS / 32), 32, 0, stream>>>(
        ucv, W_s, b_s, nullptr, sbuf, NS, DTOT, DTOT, NS, NS);

    // 4) invA (single 16x16 Gauss-Jordan)
    inv16_kernel<<<1, 256, 0, stream>>>(A, invA);

    // 5) per-position expm + discretization
    expm_kernel<<<NPOS, 32, 0, stream>>>(sbuf, A, invA, Ascl, Pmat);

    // 6) sequential state scan -> y (async double-buffered matrices)
    scan_kernel<<<dim3(DTOT / 16, BATCH), 32, 0, stream>>>(
        Ascl, Pmat, ucv, sbuf, B_param, C_param, D_param, ybuf);

    // 7) out = x + y @ W_out + b_out    (4096 x 1024 x 512)
    gemm32_wmma<<<dim3(DMODEL / 32, NPOS / 32), 32, 0, stream>>>(
        ybuf, W_out, b_out, x, out, DMODEL, DTOT, DTOT, DMODEL, DMODEL);
}